// TiltLayer_13632226197893
// MI455X (gfx1250) — compile-verified
//
#include <hip/hip_runtime.h>
#include <cmath>

#define NSZ   512
#define PLANE (512 * 512)

typedef float v2f __attribute__((ext_vector_type(2)));
typedef float v8f __attribute__((ext_vector_type(8)));

// ---------------------------------------------------------------------------
// I0 modified Bessel (Abramowitz & Stegun 9.8.1 / 9.8.2, fp32)
// ---------------------------------------------------------------------------
__device__ __forceinline__ float i0f_dev(float x) {
    float ax = fabsf(x);
    if (ax < 3.75f) {
        float t = ax / 3.75f;
        t *= t;
        return 1.0f + t * (3.5156229f + t * (3.0899424f + t * (1.2067492f +
                     t * (0.2659732f + t * (0.0360768f + t * 0.0045813f)))));
    }
    float t = 3.75f / ax;
    float p = 0.39894228f + t * (0.01328592f + t * (0.00225319f +
              t * (-0.00157565f + t * (0.00916281f + t * (-0.02057706f +
              t * (0.02635537f + t * (-0.01647633f + t * 0.00392377f)))))));
    return expf(ax) * rsqrtf(ax) * p;
}

#define BETA_F ((float)(3.141592653589793 * 2.8635642126552705)) /* pi*sqrt(8.2) */

// ---------------------------------------------------------------------------
// Build the DFT "sandwich" matrices. Each complex matrix is stored as
// [re plane][im plane] contiguous (im = re + PLANE).
//   A  [j][k] = exp(+2*pi*i*(j+256)(k+256)/512)/512          (P*W*P)
//   Bm [j][k] = exp(+2*pi*i*((j+256)%512)*k/512)/512         (P*W)
//   BmT[j][k] = Bm[k][j]
// ---------------------------------------------------------------------------
__global__ void build_mats(float* __restrict__ Amat, float* __restrict__ Bm,
                           float* __restrict__ BmT) {
    int idx = blockIdx.x * blockDim.x + threadIdx.x;
    int j = idx >> 9;
    int k = idx & 511;
    const float invN = 1.0f / 512.0f;
    const float w0   = (float)(2.0 * 3.141592653589793 / 512.0);

    int phA = ((j + 256) * (k + 256)) & 511;
    float sA, cA;
    sincosf(w0 * (float)phA, &sA, &cA);
    Amat[idx]         = cA * invN;
    Amat[idx + PLANE] = sA * invN;

    int jp  = (j + 256) & 511;
    int phB = (jp * k) & 511;
    float sB, cB;
    sincosf(w0 * (float)phB, &sB, &cB);
    Bm[idx]         = cB * invN;
    Bm[idx + PLANE] = sB * invN;

    int kp  = (k + 256) & 511;
    int phT = (kp * j) & 511;
    float sT, cT;
    sincosf(w0 * (float)phT, &sT, &cT);
    BmT[idx]         = cT * invN;
    BmT[idx + PLANE] = sT * invN;
}

// ---------------------------------------------------------------------------
// Pack the separate real/imag inputs into [re plane][im plane] per batch
// (batch stride 2*PLANE) so the GEMM can use compile-time im-plane offsets.
// ---------------------------------------------------------------------------
__global__ void pack_field(const float* __restrict__ re, const float* __restrict__ im,
                           float* __restrict__ X) {
    size_t i = (size_t)blockIdx.x * blockDim.x + threadIdx.x;  // over 4*PLANE
    size_t b = i >> 18;
    size_t o = i & (PLANE - 1);
    X[b * 2 * PLANE + o]         = re[i];
    X[b * 2 * PLANE + PLANE + o] = im[i];
}

// ---------------------------------------------------------------------------
// Complex GEMM via V_WMMA_F32_16X16X4_F32, register-blocked 32x32 per wave.
// C = A * B, 512x512x512 complex; operands are [re][im] plane pairs so the
// imaginary part is a constant +PLANE (1 MB) instruction offset.
// 2x2 grid of 16x16 WMMA tiles x 4 complex chains = 16 independent chains.
// __launch_bounds__(128,1): allow full per-wave VGPR budget (no acc spills).
// blockDim=128 (4 waves), grid=(4, 16, B).
// ---------------------------------------------------------------------------
__global__ __launch_bounds__(128, 1)
void cgemm_wmma(const float* __restrict__ A, const float* __restrict__ B,
                float* __restrict__ C,
                long long aStride, long long bStride, long long cStride) {
    A += (long long)blockIdx.z * aStride;
    B += (long long)blockIdx.z * bStride;
    C += (long long)blockIdx.z * cStride;

    const int lane = threadIdx.x & 31;
    const int wave = threadIdx.x >> 5;
    const int m = lane & 15;      // row (A) / col (B,C) within a 16x16 tile
    const int g = lane >> 4;      // half-wave group

    const int row0 = blockIdx.y * 32;
    const int col0 = (blockIdx.x * 4 + wave) * 32;

    const v8f vz = {0.f,0.f,0.f,0.f,0.f,0.f,0.f,0.f};
    v8f accRR[2][2], accII[2][2], accRI[2][2], accIR[2][2];
#pragma unroll
    for (int mt = 0; mt < 2; ++mt)
#pragma unroll
        for (int nt = 0; nt < 2; ++nt) {
            accRR[mt][nt] = vz; accII[mt][nt] = vz;
            accRI[mt][nt] = vz; accIR[mt][nt] = vz;
        }

    // A fragment: lane (g,m) holds A[row][k+2g], A[row][k+2g+1]
    // B fragment: lane (g,m) holds B[k+2g][col], B[k+2g+1][col]
    const float* aRow[2];
    const float* bCol[2];
#pragma unroll
    for (int mt = 0; mt < 2; ++mt)
        aRow[mt] = A + (size_t)(row0 + 16 * mt + m) * NSZ + 2 * g;
#pragma unroll
    for (int nt = 0; nt < 2; ++nt)
        bCol[nt] = B + (size_t)(2 * g) * NSZ + col0 + 16 * nt + m;

#pragma unroll 2
    for (int k = 0; k < NSZ; k += 4) {
        v2f aRe[2], aIm[2], bRe[2], bIm[2];
#pragma unroll
        for (int mt = 0; mt < 2; ++mt) {
            aRe[mt] = *(const v2f*)(aRow[mt]);          // offset 0
            aIm[mt] = *(const v2f*)(aRow[mt] + PLANE);  // offset +1MB (immediate)
        }
#pragma unroll
        for (int nt = 0; nt < 2; ++nt) {
            bRe[nt].x = bCol[nt][0];
            bRe[nt].y = bCol[nt][NSZ];
            bIm[nt].x = bCol[nt][PLANE];
            bIm[nt].y = bCol[nt][PLANE + NSZ];
        }
#pragma unroll
        for (int mt = 0; mt < 2; ++mt)
#pragma unroll
            for (int nt = 0; nt < 2; ++nt) {
                accRR[mt][nt] = __builtin_amdgcn_wmma_f32_16x16x4_f32(false, aRe[mt], false, bRe[nt], (short)0, accRR[mt][nt], false, false);
                accII[mt][nt] = __builtin_amdgcn_wmma_f32_16x16x4_f32(false, aIm[mt], false, bIm[nt], (short)0, accII[mt][nt], false, false);
                accRI[mt][nt] = __builtin_amdgcn_wmma_f32_16x16x4_f32(false, aRe[mt], false, bIm[nt], (short)0, accRI[mt][nt], false, false);
                accIR[mt][nt] = __builtin_amdgcn_wmma_f32_16x16x4_f32(false, aIm[mt], false, bRe[nt], (short)0, accIR[mt][nt], false, false);
            }
#pragma unroll
        for (int mt = 0; mt < 2; ++mt) aRow[mt] += 4;
#pragma unroll
        for (int nt = 0; nt < 2; ++nt) bCol[nt] += 4 * NSZ;
    }

    // C layout per 16x16 tile: VGPR r -> row r + 8*g, col = m
#pragma unroll
    for (int mt = 0; mt < 2; ++mt)
#pragma unroll
        for (int nt = 0; nt < 2; ++nt)
#pragma unroll
            for (int r = 0; r < 8; ++r) {
                const size_t o = (size_t)(row0 + 16 * mt + r + 8 * g) * NSZ
                               + col0 + 16 * nt + m;
                C[o]         = accRR[mt][nt][r] - accII[mt][nt][r];
                C[o + PLANE] = accRI[mt][nt][r] + accIR[mt][nt][r];
            }
}

// ---------------------------------------------------------------------------
__global__ void zero_buf(float* __restrict__ p, size_t n) {
    size_t i = (size_t)blockIdx.x * blockDim.x + threadIdx.x;
    if (i < n) p[i] = 0.0f;
}

// ---------------------------------------------------------------------------
// Kaiser-Bessel 1D kernel: indices (mod N, non-negative) + weights
// ---------------------------------------------------------------------------
__device__ __forceinline__ void kb1d(double omega, int* idx, float* w) {
    const double PI = 3.14159265358979323846;
    double gc   = omega * (512.0 / (2.0 * PI));
    double base = floor(gc);
#pragma unroll
    for (int t = 0; t < 6; ++t) {
        double id   = base + (double)(t - 2);
        double dist = gc - id;
        float  d3   = (float)(dist * (1.0 / 3.0));
        float  arg  = 1.0f - d3 * d3;
        float  wv   = 0.0f;
        if (arg > 0.0f) wv = i0f_dev(BETA_F * sqrtf(fmaxf(arg, 1e-12f)));
        w[t] = wv;
        int ii = (int)id;
        ii %= 512;
        if (ii < 0) ii += 512;
        idx[t] = ii;
    }
}

// ---------------------------------------------------------------------------
// Trajectory + gridding: one thread per sample point, scatter-add 6x6 taps
// for all 4 batches with native fp32 global atomics.
// Planes are [re][im] pairs, batch stride 2*PLANE.
// ---------------------------------------------------------------------------
__global__ void grid_kb(const float* __restrict__ freq,
                        const float* __restrict__ ang1, const float* __restrict__ ang2,
                        float* __restrict__ grd) {
    int i = blockIdx.x * blockDim.x + threadIdx.x;
    if (i >= PLANE) return;

    // GRID uses a full flip of the flattened (row, col) index
    int r    = PLANE - 1 - i;
    int rowi = r >> 9;
    int coli = r & 511;

    const double PI      = 3.14159265358979323846;
    const double WL      = 6.33e-07;
    const double SPACING = 3.45e-06;
    const double MAXK    = 1.0 / (2.0 * SPACING);
    const double MAXW    = (1.0 / WL) / MAXK * PI;

    double g0 = ((double)coli * (2.0 / 512.0) - 1.0) * PI;   // x
    double g1 = ((double)rowi * (2.0 / 512.0) - 1.0) * PI;   // y
    double kx = g0 / PI * MAXK;
    double ky = g1 / PI * MAXK;
    double Wv = sqrt(1.0 / (WL * WL) - kx * kx - ky * ky) / MAXK * PI;

    double a1 = (double)ang1[0];
    double a2 = (double)ang2[0];
    double s1 = sin(a1), c1 = cos(a1), s2 = sin(a2), c2 = cos(a2);

    double p0 = c1 * g0 + s1 * s2 * g1 + s1 * c2 * Wv
              - MAXW * s1 * c2 - s1 * s2 * PI;
    double p1 = c2 * g1 - s2 * Wv + MAXW * s2;
    p0 /= c1;
    p1 /= c2;

    int iy[6], ix[6];
    float wy[6], wx[6];
    kb1d(p1, iy, wy);   // traj[0] = p1 -> y axis
    kb1d(p0, ix, wx);   // traj[1] = p0 -> x axis

    float vr[4], vi[4];
#pragma unroll
    for (int b = 0; b < 4; ++b) {
        vr[b] = freq[(size_t)b * 2 * PLANE + i];
        vi[b] = freq[(size_t)b * 2 * PLANE + PLANE + i];
    }

#pragma unroll
    for (int ty = 0; ty < 6; ++ty) {
#pragma unroll
        for (int tx = 0; tx < 6; ++tx) {
            float w    = wy[ty] * wx[tx];
            int   addr = iy[ty] * 512 + ix[tx];
#pragma unroll
            for (int b = 0; b < 4; ++b) {
                float* gb = grd + (size_t)b * 2 * PLANE + addr;
                unsafeAtomicAdd(gb,         vr[b] * w);
                unsafeAtomicAdd(gb + PLANE, vi[b] * w);
            }
        }
    }
}

// ---------------------------------------------------------------------------
// out = img * N*N / (apod[y] * apod[x]), interleaved complex64
// ---------------------------------------------------------------------------
__device__ __forceinline__ float apodf(int n_) {
    float n = (float)(n_ - 256);
    float u = (float)(3.141592653589793) * 6.0f * n * (1.0f / 512.0f);
    float t = BETA_F * BETA_F - u * u;
    float s = sqrtf(fabsf(t)) + 1e-12f;
    return (t > 0.0f) ? (sinhf(s) / s) : (sinf(s) / s);
}

__global__ void finalize_k(const float* __restrict__ C2, float* __restrict__ out) {
    int i = blockIdx.x * blockDim.x + threadIdx.x;
    if (i >= PLANE) return;
    int b = blockIdx.y;
    int y = i >> 9;
    int x = i & 511;
    float sc = (512.0f * 512.0f) / (apodf(y) * apodf(x));
    const float* Cb = C2 + (size_t)b * 2 * PLANE;
    size_t o = (size_t)b * PLANE + i;
    out[2 * o]     = Cb[i] * sc;
    out[2 * o + 1] = Cb[i + PLANE] * sc;
}

// ---------------------------------------------------------------------------
extern "C" void kernel_launch(void* const* d_in, const int* in_sizes, int n_in,
                              void* d_out, int out_size, void* d_ws, size_t ws_size,
                              hipStream_t stream) {
    (void)in_sizes; (void)n_in; (void)out_size; (void)ws_size;

    const float* fieldRe = (const float*)d_in[0];   // (B,N,N) f32
    const float* fieldIm = (const float*)d_in[1];   // (B,N,N) f32
    const float* ang1    = (const float*)d_in[2];   // (1,)   f32
    const float* ang2    = (const float*)d_in[3];   // (1,)   f32
    float*       out     = (float*)d_out;           // (B,1,N,N) complex64 interleaved

    const size_t P  = (size_t)PLANE;
    const long long S = 2 * (long long)PLANE;       // per-batch complex stride
    float* ws   = (float*)d_ws;
    float* Amat = ws;               // 2P  : P*W*P
    float* Bm   = Amat + 2 * P;     // 2P  : P*W
    float* BmT  = Bm   + 2 * P;     // 2P  : (P*W)^T
    float* X    = BmT  + 2 * P;     // 8P  : packed input field
    float* bufA = X    + 8 * P;     // 8P  : T1 / grid / C2
    float* bufB = bufA + 8 * P;     // 8P  : freq / C1

    dim3 gemmBlk(128);
    dim3 gemmGrd(4, 16, 4);   // 32x32 complex tile per wave, 4 waves/block

    // DFT matrices + input packing
    build_mats<<<PLANE / 256, 256, 0, stream>>>(Amat, Bm, BmT);
    pack_field<<<(unsigned)((4 * P) / 256), 256, 0, stream>>>(fieldRe, fieldIm, X);

    // Stage 1: freq = A * X * A  (= ifftshift(ifft2(ifftshift(field))))
    cgemm_wmma<<<gemmGrd, gemmBlk, 0, stream>>>(Amat, X,    bufA, 0, S, S);
    cgemm_wmma<<<gemmGrd, gemmBlk, 0, stream>>>(bufA, Amat, bufB, S, 0, S);

    // Stage 2: NUFFT gridding -> grid in bufA (zeroed first; 8P contiguous)
    zero_buf<<<(unsigned)((8 * P + 255) / 256), 256, 0, stream>>>(bufA, 8 * P);
    grid_kb<<<(PLANE + 255) / 256, 256, 0, stream>>>(bufB, ang1, ang2, bufA);

    // Stage 3: img = (P*W) * grid * (P*W)^T  (= fftshift(ifft2(grid)))
    cgemm_wmma<<<gemmGrd, gemmBlk, 0, stream>>>(Bm,   bufA, bufB, 0, S, S);
    cgemm_wmma<<<gemmGrd, gemmBlk, 0, stream>>>(bufB, BmT,  bufA, S, 0, S);

    // Stage 4: scale + apodization -> interleaved complex64 output
    dim3 finGrd((PLANE + 255) / 256, 4);
    finalize_k<<<finGrd, 256, 0, stream>>>(bufA, out);
}